// FGL_1443109012165
// MI455X (gfx1250) — compile-verified
//
#include <hip/hip_runtime.h>

typedef __attribute__((ext_vector_type(2))) float v2f;
typedef __attribute__((ext_vector_type(8))) float v8f;

constexpr int B    = 32;
constexpr int INC  = 32;
constexpr int OUTC = 64;
constexpr int INN  = 65536;
constexpr int OUTN = 8192;
constexpr int DEG  = 16;

constexpr int NT      = 16;        // output-column tile per workgroup
constexpr int THREADS = 128;       // 4 waves (wave32)
constexpr int CP      = INC + 2;   // padded K-stride (34) -> conflict-free LDS

// Fused: gather + hadamard + max-reduce -> LDS -> f32 WMMA channel GEMM -> +bias
__global__ __launch_bounds__(THREADS)
void fgl_fused_kernel(const float* __restrict__ x,    // (B, INC, INN)
                      const int*   __restrict__ idx,  // (OUTN, DEG)
                      const float* __restrict__ nf,   // (INC, INN)
                      const float* __restrict__ ft,   // (INC, OUTC)
                      const float* __restrict__ bias, // (OUTC, OUTN)
                      float*       __restrict__ out)  // (B, OUTC, OUTN)
{
    __shared__ alignas(16) int   s_idx[NT * DEG];   // 256 ints
    __shared__ alignas(16) float s_red[NT * CP];    // red^T[nl][c], K-contiguous
    __shared__ alignas(16) float s_ft [OUTC * CP];  // ft^T[k][c],  K-contiguous

    const int tid   = threadIdx.x;
    const int ntile = blockIdx.x;          // [0, OUTN/NT)
    const int b     = blockIdx.y;          // [0, B)

    // ---- stage idx tile (vectorized) + ft transposed into LDS ----
    {
        const int4* __restrict__ idx4 = (const int4*)(idx + ntile * (NT * DEG));
        int4* s_idx4 = (int4*)s_idx;
        if (tid < (NT * DEG) / 4)
            s_idx4[tid] = idx4[tid];
    }
    for (int i = tid; i < INC * OUTC; i += THREADS) {
        const int c = i >> 6;              // ft row-major (INC, OUTC)
        const int k = i & 63;
        s_ft[k * CP + c] = ft[i];          // transpose: K (=c) contiguous per output row
    }
    __syncthreads();

    // ---- phase 1: gather, hadamard, max over DEG neighbors ----
    const size_t xb = (size_t)b * INC * INN;
    for (int p = tid; p < INC * NT; p += THREADS) {
        const int c  = p >> 4;
        const int nl = p & 15;
        const float* __restrict__ xc = x  + xb + (size_t)c * INN;
        const float* __restrict__ nc = nf +      (size_t)c * INN;
        float m = -__builtin_inff();
        #pragma unroll
        for (int d = 0; d < DEG; ++d) {
            const int j = s_idx[nl * DEG + d];
            m = fmaxf(m, xc[j] * nc[j]);
        }
        s_red[nl * CP + c] = m;            // transposed: K (=c) contiguous
    }
    __syncthreads();

    // ---- phase 2: each wave computes a 16x16 tile of y (k-tile x n-tile), K=INC=32 ----
    const int wave = tid >> 5;             // 0..3 -> k-tile over OUTC
    const int lane = tid & 31;
    const int half = lane >> 4;            // selects K sub-pair per ISA A/B layout
    const int col  = lane & 15;            // A: M index; B,C,D: N index
    const int krow = wave * 16 + col;      // output channel for A-matrix loads

    v8f acc = {0.f, 0.f, 0.f, 0.f, 0.f, 0.f, 0.f, 0.f};
    #pragma unroll
    for (int kc = 0; kc < INC / 4; ++kc) {
        const int c0 = kc * 4 + 2 * half;                       // even -> 8B aligned
        // A (16x4, MxK): A[M=col][K=c0+v] = ft^T[krow][c0+v]   (one ds_load_b64)
        const v2f a  = *(const v2f*)&s_ft [krow * CP + c0];
        // B (4x16, KxN): B[K=c0+v][N=col] = red^T[col][c0+v]   (one ds_load_b64)
        const v2f bm = *(const v2f*)&s_red[col  * CP + c0];
        acc = __builtin_amdgcn_wmma_f32_16x16x4_f32(
            /*neg_a=*/false, a, /*neg_b=*/false, bm,
            /*c_mod=*/(short)0, acc, /*reuse_a=*/false, /*reuse_b=*/false);
    }

    // ---- phase 3: bias epilogue + store (C/D layout: M = v + 8*half, N = col) ----
    const int n = ntile * NT + col;
    const size_t ob = (size_t)b * OUTC * OUTN;
    #pragma unroll
    for (int v = 0; v < 8; ++v) {
        const int k = wave * 16 + half * 8 + v;
        out[ob + (size_t)k * OUTN + n] = acc[v] + bias[k * OUTN + n];
    }
}

extern "C" void kernel_launch(void* const* d_in, const int* in_sizes, int n_in,
                              void* d_out, int out_size, void* d_ws, size_t ws_size,
                              hipStream_t stream) {
    const float* x    = (const float*)d_in[0];
    const int*   idx  = (const int*)  d_in[1];
    const float* nf   = (const float*)d_in[2];
    const float* ft   = (const float*)d_in[3];
    const float* bias = (const float*)d_in[4];
    float*       out  = (float*)d_out;

    dim3 grid(OUTN / NT, B);
    fgl_fused_kernel<<<grid, THREADS, 0, stream>>>(x, idx, nf, ft, bias, out);
}